// _CrossScaleNeighborhoodExtractor_46110768890409
// MI455X (gfx1250) — compile-verified
//
#include <hip/hip_runtime.h>

// Problem constants from setup_inputs()/reference:
#define BATCH   2
#define CHN     128          // = HEADS*HD
#define HEADS   8
#define HD      16
#define HCC     48           // coarse H
#define WCC     48           // coarse W
#define KWIN    25           // 5x5 window offsets
#define FH      96           // fine (query) H
#define FW      96           // fine (query) W
#define PLANE_C (HCC * WCC)  // 2304 floats
#define PLANE_F (FH * FW)    // 9216 floats

#define VOFF 58982400u       // elements: B*HEADS*KWIN*HD*FH*FW
#define MOFF 117964800u      // 2*VOFF

typedef float v4f __attribute__((ext_vector_type(4)));
typedef __attribute__((address_space(3))) float lds_f;

// Raw LDS byte offset (what GLOBAL_LOAD_ASYNC_TO_LDS wants in VDST)
__device__ __forceinline__ unsigned lds_off(const float* p) {
  return (unsigned)(unsigned long long)(lds_f*)p;
}

// CDNA5 async global->LDS copy, 16 bytes, tracked by ASYNCcnt.
__device__ __forceinline__ void async_copy_b128(unsigned ldsaddr, const float* gaddr) {
  asm volatile("global_load_async_to_lds_b128 %0, %1, off"
               :: "v"(ldsaddr), "v"(gaddr)
               : "memory");
}

__device__ __forceinline__ void wait_async_all() {
  asm volatile("s_wait_asynccnt 0" ::: "memory");
}

// One block = one (b, channel) coarse plane x 5 window offsets.
// Stage k/v planes in LDS via async copies, then stream NT float4 stores.
__global__ __launch_bounds__(256)
void csne_kv_kernel(const float* __restrict__ kin,
                    const float* __restrict__ vin,
                    float* __restrict__ out) {
  __shared__ __align__(16) float sk[PLANE_C];
  __shared__ __align__(16) float sv[PLANE_C];

  const int tid   = threadIdx.x;
  const int plane = blockIdx.x;            // b*CHN + ch
  const int b     = plane / CHN;
  const int ch    = plane - b * CHN;
  const int h     = ch / HD;
  const int d     = ch - h * HD;
  const int kog   = blockIdx.y;            // offset group: ko = 5*kog + q

  const float* gk = kin + (size_t)plane * PLANE_C;
  const float* gv = vin + (size_t)plane * PLANE_C;

  // Stage both 9KB coarse planes into LDS (576 b128 async copies each).
  for (int i = tid; i < PLANE_C / 4; i += 256) {
    async_copy_b128(lds_off(&sk[i * 4]), gk + i * 4);
    async_copy_b128(lds_off(&sv[i * 4]), gv + i * 4);
  }
  wait_async_all();
  __syncthreads();

  for (int q = 0; q < 5; ++q) {
    const int ko = kog * 5 + q;
    const int dy = ko / 5 - 2;
    const int dx = ko % 5 - 2;
    const size_t pbase =
        ((size_t)((b * HEADS + h) * KWIN + ko) * HD + d) * PLANE_F;
    float* __restrict__ ok = out + pbase;
    float* __restrict__ ov = out + VOFF + pbase;

    // item = (coarse row yc, fine float4-group xg). Each item emits 4 NT
    // b128 stores (2 tensors x 2 identical fine rows).
    for (int item = tid; item < (FH / 2) * (FW / 4); item += 256) {
      const int yc = item / (FW / 4);
      const int xg = item - yc * (FW / 4);
      const int ys = yc + dy;              // shifted coarse row
      const int x0 = 2 * xg + dx;          // shifted coarse cols
      const int x1 = x0 + 1;
      const bool vy = ((unsigned)ys < HCC);
      const bool c0 = vy && ((unsigned)x0 < WCC);
      const bool c1 = vy && ((unsigned)x1 < WCC);
      const int row = ys * WCC;
      const float k0 = c0 ? sk[row + x0] : 0.0f;
      const float k1 = c1 ? sk[row + x1] : 0.0f;
      const float f0 = c0 ? sv[row + x0] : 0.0f;
      const float f1 = c1 ? sv[row + x1] : 0.0f;
      const v4f kq = {k0, k0, k1, k1};
      const v4f vq = {f0, f0, f1, f1};
      const int o = (2 * yc) * FW + xg * 4;
      __builtin_nontemporal_store(kq, (v4f*)(ok + o));
      __builtin_nontemporal_store(kq, (v4f*)(ok + o + FW));
      __builtin_nontemporal_store(vq, (v4f*)(ov + o));
      __builtin_nontemporal_store(vq, (v4f*)(ov + o + FW));
    }
  }
}

// Mask planes [B,HEADS,KWIN,96,96]: pure arithmetic, no loads.
__global__ __launch_bounds__(256)
void csne_mask_kernel(float* __restrict__ out) {
  const int p  = blockIdx.x;               // (b*HEADS+h)*KWIN + ko
  const int ko = p % KWIN;
  const int dy = ko / 5 - 2;
  const int dx = ko % 5 - 2;
  float* __restrict__ om = out + MOFF + (size_t)p * PLANE_F;

  for (int item = threadIdx.x; item < (FH / 2) * (FW / 4); item += 256) {
    const int yc = item / (FW / 4);
    const int xg = item - yc * (FW / 4);
    const int ys = yc + dy;
    const int x0 = 2 * xg + dx;
    const int x1 = x0 + 1;
    const bool vy = ((unsigned)ys < HCC);
    const float m0 = (vy && ((unsigned)x0 < WCC)) ? 1.0f : 0.0f;
    const float m1 = (vy && ((unsigned)x1 < WCC)) ? 1.0f : 0.0f;
    const v4f mq = {m0, m0, m1, m1};
    const int o = (2 * yc) * FW + xg * 4;
    __builtin_nontemporal_store(mq, (v4f*)(om + o));
    __builtin_nontemporal_store(mq, (v4f*)(om + o + FW));
  }
}

extern "C" void kernel_launch(void* const* d_in, const int* in_sizes, int n_in,
                              void* d_out, int out_size, void* d_ws, size_t ws_size,
                              hipStream_t stream) {
  const float* k = (const float*)d_in[0];
  const float* v = (const float*)d_in[1];
  float* out = (float*)d_out;

  dim3 gridKV(BATCH * CHN, 5);             // 1280 blocks
  csne_kv_kernel<<<gridKV, 256, 0, stream>>>(k, v, out);

  csne_mask_kernel<<<BATCH * HEADS * KWIN, 256, 0, stream>>>(out);
}